// GNN_Parameterization_24567212933210
// MI455X (gfx1250) — compile-verified
//
#include <hip/hip_runtime.h>

// Problem constants (fixed by the reference)
#define NGRAPH 32
#define NPG    2048
#define NN     (NGRAPH * NPG)     // 65536 nodes
#define NE     (NN * 16)          // 1048576 edges
#define FIN    128
#define HD     128
#define H2D    32
#define NC     10
#define KKEEP  1024
#define N2     (NGRAPH * KKEEP)   // 32768 kept nodes

typedef float v2f __attribute__((ext_vector_type(2)));
typedef float v8f __attribute__((ext_vector_type(8)));

// ---------------------------------------------------------------- utilities
__global__ void k_zero_i(int* __restrict__ p, int n) {
    int i = blockIdx.x * blockDim.x + threadIdx.x;
    if (i < n) p[i] = 0;
}

// ---------------------------------------------------------------- GEMM1: xw1 = (x * t_x) @ W1   [65536x128 @ 128x128]
// One wave = 16-row x 128-col output strip. W1 staged in LDS.
// fp32 WMMA 16x16x4: A frag (16x4) = v2f, lanes 16-31 hold K+2; B frag (4x16) mirrors C layout.
__global__ __launch_bounds__(256) void k_gemm1(const float* __restrict__ x,
                                               const float* __restrict__ tx,
                                               const float* __restrict__ W1,
                                               float* __restrict__ xw) {
    __shared__ __align__(16) float sW[HD * HD];   // 64 KB
    int tid = threadIdx.x;
    for (int i = tid * 4; i < HD * HD; i += 256 * 4)
        *(float4*)(sW + i) = *(const float4*)(W1 + i);
    __syncthreads();

    int wave = tid >> 5, lane = tid & 31;
    int Mbase = (blockIdx.x * 8 + wave) * 16;
    int m     = Mbase + (lane & 15);
    int khalf = (lane >> 4) * 2;     // lanes 16-31 carry K+2, K+3
    int n     = lane & 15;

    const float* xr = x  + (size_t)m * FIN;
    const float* tr = tx + (size_t)m * FIN;

    v8f acc[8] = {};
    for (int ks = 0; ks < 32; ++ks) {
        int k0 = ks * 4 + khalf;
        v2f a;
        a.x = xr[k0]     * tr[k0];       // fuse LambdaLayer multiply into A load
        a.y = xr[k0 + 1] * tr[k0 + 1];
#pragma unroll
        for (int t = 0; t < 8; ++t) {
            v2f b;
            b.x = sW[k0 * HD + t * 16 + n];
            b.y = sW[(k0 + 1) * HD + t * 16 + n];
            acc[t] = __builtin_amdgcn_wmma_f32_16x16x4_f32(
                false, a, false, b, (short)0, acc[t], false, false);
        }
    }
    int rowoff = (lane >> 4) * 8;    // C/D: lanes 16-31 hold rows M+8
#pragma unroll
    for (int t = 0; t < 8; ++t)
#pragma unroll
        for (int r = 0; r < 8; ++r)
            xw[(size_t)(Mbase + rowoff + r) * HD + t * 16 + n] = acc[t][r];
}

// ---------------------------------------------------------------- CSR construction
__global__ void k_count1(const int* __restrict__ dst, int* __restrict__ cnt) {
    int e = blockIdx.x * blockDim.x + threadIdx.x;
    if (e < NE) atomicAdd(&cnt[dst[e]], 1);
}

__global__ void k_count2(const int* __restrict__ src, const int* __restrict__ dst,
                         const int* __restrict__ newid, int* __restrict__ cnt) {
    int e = blockIdx.x * blockDim.x + threadIdx.x;
    if (e >= NE) return;
    int s2 = newid[src[e]], d2 = newid[dst[e]];
    if (s2 >= 0 && d2 >= 0) atomicAdd(&cnt[d2], 1);
}

// Single-workgroup exclusive prefix scan (n must be multiple of 1024)
__global__ __launch_bounds__(1024) void k_scan(const int* __restrict__ cnt,
                                               int* __restrict__ offs, int n) {
    __shared__ int ssum[1024];
    int t = threadIdx.x;
    int chunk = n >> 10;
    int base = t * chunk;
    int s = 0;
    for (int i = 0; i < chunk; ++i) s += cnt[base + i];
    ssum[t] = s;
    __syncthreads();
    for (int off = 1; off < 1024; off <<= 1) {   // Hillis-Steele inclusive scan
        int v = (t >= off) ? ssum[t - off] : 0;
        __syncthreads();
        ssum[t] += v;
        __syncthreads();
    }
    int run = ssum[t] - s;                        // exclusive base for this chunk
    for (int i = 0; i < chunk; ++i) { offs[base + i] = run; run += cnt[base + i]; }
    if (t == 1023) offs[n] = run;                 // total
}

__global__ void k_dinvk(const int* __restrict__ cnt, float* __restrict__ dinv, int n) {
    int i = blockIdx.x * blockDim.x + threadIdx.x;
    if (i < n) dinv[i] = rsqrtf((float)cnt[i] + 1.0f);   // +1 self loop
}

__global__ void k_fill1(const int* __restrict__ src, const int* __restrict__ dst,
                        const int* __restrict__ offs, int* __restrict__ cursor,
                        int* __restrict__ csr) {
    int e = blockIdx.x * blockDim.x + threadIdx.x;
    if (e >= NE) return;
    int d = dst[e];
    int slot = atomicAdd(&cursor[d], 1);
    csr[offs[d] + slot] = src[e];
}

__global__ void k_fill2(const int* __restrict__ src, const int* __restrict__ dst,
                        const int* __restrict__ newid, const int* __restrict__ offs,
                        int* __restrict__ cursor, int* __restrict__ csr) {
    int e = blockIdx.x * blockDim.x + threadIdx.x;
    if (e >= NE) return;
    int s2 = newid[src[e]], d2 = newid[dst[e]];
    if (s2 < 0 || d2 < 0) return;
    int slot = atomicAdd(&cursor[d2], 1);
    csr[offs[d2] + slot] = s2;
}

// ---------------------------------------------------------------- conv1 gather: one wave per dst node, 4 floats/lane.
// Fuses self-loop, bias, and the TopK pooling score (saves a 32MB re-read).
__global__ __launch_bounds__(256) void k_gather1(const int* __restrict__ offs,
                                                 const int* __restrict__ csr,
                                                 const float* __restrict__ dinv,
                                                 const float* __restrict__ xw,
                                                 const float* __restrict__ b1,
                                                 const float* __restrict__ pw,
                                                 float* __restrict__ h1,
                                                 float* __restrict__ score) {
    int node = (blockIdx.x * 256 + threadIdx.x) >> 5;
    int lane = threadIdx.x & 31;
    if (node >= NN) return;
    int beg = offs[node], end = offs[node + 1];
    float dn = dinv[node];
    float c  = dn * dn;
    float4 v = *(const float4*)(xw + (size_t)node * HD + lane * 4);
    float4 acc;
    acc.x = v.x * c; acc.y = v.y * c; acc.z = v.z * c; acc.w = v.w * c;
    for (int e = beg; e < end; ++e) {             // uniform loop (whole wave = one node)
        int s = csr[e];
        if (e + 1 < end)                          // hide pointer-chase latency
            __builtin_prefetch(&xw[(size_t)csr[e + 1] * HD], 0, 0);
        float nrm = dinv[s] * dn;
        float4 u = *(const float4*)(xw + (size_t)s * HD + lane * 4);
        acc.x += u.x * nrm; acc.y += u.y * nrm; acc.z += u.z * nrm; acc.w += u.w * nrm;
    }
    float4 bb = ((const float4*)b1)[lane];
    acc.x += bb.x; acc.y += bb.y; acc.z += bb.z; acc.w += bb.w;
    *(float4*)(h1 + (size_t)node * HD + lane * 4) = acc;

    // fused pooling score: tanh(h . w / ||w||)
    float4 w4 = ((const float4*)pw)[lane];
    float wn = w4.x * w4.x + w4.y * w4.y + w4.z * w4.z + w4.w * w4.w;
    float dv = acc.x * w4.x + acc.y * w4.y + acc.z * w4.z + acc.w * w4.w;
    for (int off = 16; off; off >>= 1) {
        wn += __shfl_xor(wn, off);
        dv += __shfl_xor(dv, off);
    }
    if (lane == 0) score[node] = tanhf(dv * rsqrtf(wn));
}

// ---------------------------------------------------------------- per-graph top-1024 via LDS bitonic sort (descending)
__global__ __launch_bounds__(1024) void k_topk(const float* __restrict__ score,
                                               int* __restrict__ newid,
                                               int* __restrict__ perm,
                                               float* __restrict__ sk_out) {
    __shared__ float skey[NPG];
    __shared__ int   sidx[NPG];
    int g = blockIdx.x, t = threadIdx.x;       // 1024 threads
    skey[t] = score[g * NPG + t];               sidx[t] = t;
    skey[t + 1024] = score[g * NPG + t + 1024]; sidx[t + 1024] = t + 1024;
    __syncthreads();
    for (int k = 2; k <= NPG; k <<= 1) {
        for (int j = k >> 1; j > 0; j >>= 1) {
            int i   = ((t & ~(j - 1)) << 1) | (t & (j - 1));
            int ixj = i | j;
            bool up = ((i & k) == 0);
            float a = skey[i], b = skey[ixj];
            bool sw = up ? (a < b) : (a > b);   // descending sort
            if (sw) {
                skey[i] = b; skey[ixj] = a;
                int tmp = sidx[i]; sidx[i] = sidx[ixj]; sidx[ixj] = tmp;
            }
            __syncthreads();
        }
    }
    // ranks 0..1023 kept, 1024..2047 dropped
    int orig = g * NPG + sidx[t];
    int nid  = g * KKEEP + t;
    perm[nid]   = orig;
    sk_out[nid] = skey[t];
    newid[orig] = nid;
    newid[g * NPG + sidx[t + 1024]] = -1;
}

// ---------------------------------------------------------------- GEMM2: xw2 = relu(h1[perm]*score) @ W2   [32768x128 @ 128x32]
__global__ __launch_bounds__(256) void k_gemm2(const float* __restrict__ h1,
                                               const int* __restrict__ perm,
                                               const float* __restrict__ sk,
                                               const float* __restrict__ W2,
                                               float* __restrict__ xw2) {
    __shared__ __align__(16) float sW[HD * H2D];   // 16 KB
    int tid = threadIdx.x;
    for (int i = tid * 4; i < HD * H2D; i += 256 * 4)
        *(float4*)(sW + i) = *(const float4*)(W2 + i);
    __syncthreads();

    int wave = tid >> 5, lane = tid & 31;
    int Mbase = (blockIdx.x * 8 + wave) * 16;
    int row   = Mbase + (lane & 15);
    int orig  = perm[row];
    float sc  = sk[row];
    const float* hr = h1 + (size_t)orig * HD;
    int khalf = (lane >> 4) * 2;
    int n     = lane & 15;

    v8f acc[2] = {};
    for (int ks = 0; ks < 32; ++ks) {
        int k0 = ks * 4 + khalf;
        v2f a;
        a.x = fmaxf(hr[k0]     * sc, 0.0f);   // relu(h * score) fused into gather
        a.y = fmaxf(hr[k0 + 1] * sc, 0.0f);
#pragma unroll
        for (int t = 0; t < 2; ++t) {
            v2f b;
            b.x = sW[k0 * H2D + t * 16 + n];
            b.y = sW[(k0 + 1) * H2D + t * 16 + n];
            acc[t] = __builtin_amdgcn_wmma_f32_16x16x4_f32(
                false, a, false, b, (short)0, acc[t], false, false);
        }
    }
    int rowoff = (lane >> 4) * 8;
#pragma unroll
    for (int t = 0; t < 2; ++t)
#pragma unroll
        for (int r = 0; r < 8; ++r)
            xw2[(size_t)(Mbase + rowoff + r) * H2D + t * 16 + n] = acc[t][r];
}

// ---------------------------------------------------------------- conv2 gather: one wave per kept node, 1 float/lane.
// Fuses self-loop, bias, relu.
__global__ __launch_bounds__(256) void k_gather2(const int* __restrict__ offs,
                                                 const int* __restrict__ csr,
                                                 const float* __restrict__ dinv2,
                                                 const float* __restrict__ xw2,
                                                 const float* __restrict__ b2,
                                                 float* __restrict__ h2) {
    int node = (blockIdx.x * 256 + threadIdx.x) >> 5;
    int lane = threadIdx.x & 31;
    if (node >= N2) return;
    int beg = offs[node], end = offs[node + 1];
    float dn = dinv2[node];
    float acc = xw2[(size_t)node * H2D + lane] * dn * dn;
    for (int e = beg; e < end; ++e) {
        int s = csr[e];
        if (e + 1 < end)
            __builtin_prefetch(&xw2[(size_t)csr[e + 1] * H2D], 0, 0);
        acc += xw2[(size_t)s * H2D + lane] * (dinv2[s] * dn);
    }
    acc += b2[lane];
    h2[(size_t)node * H2D + lane] = fmaxf(acc, 0.0f);
}

// ---------------------------------------------------------------- mean-pool + classifier (32x32 @ 32x10)
__global__ __launch_bounds__(256) void k_pool(const float* __restrict__ h2,
                                              const float* __restrict__ Wl,
                                              const float* __restrict__ bl,
                                              float* __restrict__ out) {
    int g = blockIdx.x, t = threadIdx.x;     // 256 threads: 8 row-groups x 32 cols
    int c = t & 31, r0 = t >> 5;
    float sum = 0.0f;
    for (int r = r0; r < KKEEP; r += 8)
        sum += h2[((size_t)g * KKEEP + r) * H2D + c];
    __shared__ float red[256];
    __shared__ float pooled[H2D];
    red[t] = sum; __syncthreads();
    if (t < 128) red[t] += red[t + 128]; __syncthreads();
    if (t < 64)  red[t] += red[t + 64];  __syncthreads();
    if (t < 32)  red[t] += red[t + 32];  __syncthreads();
    if (t < 32)  pooled[t] = red[t] * (1.0f / (float)KKEEP);
    __syncthreads();
    if (t < NC) {
        float acc = bl[t];
#pragma unroll
        for (int k = 0; k < H2D; ++k) acc += pooled[k] * Wl[k * NC + t];
        out[g * NC + t] = acc;
    }
}

// ---------------------------------------------------------------- launch
extern "C" void kernel_launch(void* const* d_in, const int* in_sizes, int n_in,
                              void* d_out, int out_size, void* d_ws, size_t ws_size,
                              hipStream_t stream) {
    const float* x   = (const float*)d_in[0];
    const float* tx  = (const float*)d_in[1];
    const int*   ei  = (const int*)d_in[2];
    // d_in[3] = batch (unused: uniform graph sizes)
    const float* W1  = (const float*)d_in[4];
    const float* b1  = (const float*)d_in[5];
    const float* pw  = (const float*)d_in[6];
    const float* W2  = (const float*)d_in[7];
    const float* b2  = (const float*)d_in[8];
    const float* Wl  = (const float*)d_in[9];
    const float* bl  = (const float*)d_in[10];
    float* out = (float*)d_out;
    const int* src = ei;
    const int* dst = ei + NE;

    // Workspace layout (~90 MB). Big vector-accessed arrays first (16B aligned).
    float* ws = (float*)d_ws;
    size_t off = 0;
    float* xw1   = ws + off; off += (size_t)NN * HD;   // 32 MB
    float* h1    = ws + off; off += (size_t)NN * HD;   // 32 MB
    float* xw2   = ws + off; off += (size_t)N2 * H2D;  // 4 MB
    float* h2    = ws + off; off += (size_t)N2 * H2D;  // 4 MB
    float* dinv1 = ws + off; off += NN;
    float* dinv2 = ws + off; off += N2;
    float* score = ws + off; off += NN;
    float* sk    = ws + off; off += N2;
    int* newid   = (int*)(ws + off); off += NN;
    int* perm    = (int*)(ws + off); off += N2;
    int* cnt1    = (int*)(ws + off); off += NN;
    int* cursor1 = (int*)(ws + off); off += NN;
    int* cnt2    = (int*)(ws + off); off += N2;
    int* cursor2 = (int*)(ws + off); off += N2;
    int* csr1    = (int*)(ws + off); off += NE;        // 4 MB
    int* csr2    = (int*)(ws + off); off += NE;        // 4 MB
    int* offs1   = (int*)(ws + off); off += NN + 1;
    int* offs2   = (int*)(ws + off); off += N2 + 1;

    // zero counters
    k_zero_i<<<(NN + 255) / 256, 256, 0, stream>>>(cnt1, NN);
    k_zero_i<<<(NN + 255) / 256, 256, 0, stream>>>(cursor1, NN);
    k_zero_i<<<(N2 + 255) / 256, 256, 0, stream>>>(cnt2, N2);
    k_zero_i<<<(N2 + 255) / 256, 256, 0, stream>>>(cursor2, N2);

    // conv1: GEMM + CSR build + gather (fused bias + pooling score)
    k_gemm1<<<NN / 128, 256, 0, stream>>>(x, tx, W1, xw1);
    k_count1<<<NE / 256, 256, 0, stream>>>(dst, cnt1);
    k_scan<<<1, 1024, 0, stream>>>(cnt1, offs1, NN);
    k_dinvk<<<NN / 256, 256, 0, stream>>>(cnt1, dinv1, NN);
    k_fill1<<<NE / 256, 256, 0, stream>>>(src, dst, offs1, cursor1, csr1);
    k_gather1<<<NN / 8, 256, 0, stream>>>(offs1, csr1, dinv1, xw1, b1, pw, h1, score);

    // top-k pooling
    k_topk<<<NGRAPH, 1024, 0, stream>>>(score, newid, perm, sk);

    // conv2: gathered GEMM + CSR build on remapped graph + gather (fused bias + relu)
    k_gemm2<<<N2 / 128, 256, 0, stream>>>(h1, perm, sk, W2, xw2);
    k_count2<<<NE / 256, 256, 0, stream>>>(src, dst, newid, cnt2);
    k_scan<<<1, 1024, 0, stream>>>(cnt2, offs2, N2);
    k_dinvk<<<N2 / 256, 256, 0, stream>>>(cnt2, dinv2, N2);
    k_fill2<<<NE / 256, 256, 0, stream>>>(src, dst, newid, offs2, cursor2, csr2);
    k_gather2<<<N2 / 8, 256, 0, stream>>>(offs2, csr2, dinv2, xw2, b2, h2);

    // mean pool + linear
    k_pool<<<NGRAPH, 256, 0, stream>>>(h2, Wl, bl, out);
}